// GINNA_76699525972535
// MI455X (gfx1250) — compile-verified
//
#include <hip/hip_runtime.h>
#include <math.h>

// ---------------------------------------------------------------------------
// GIN forward for MI455X (gfx1250, wave32).
//  - scatter-add aggregation with f32 global atomics (L2-resident working set)
//  - fp32 WMMA (V_WMMA_F32_16X16X4_F32) GEMMs with fused bias+BN+LeakyReLU
//  - W tile staged TRANSPOSED in LDS -> B fragment = single ds_load_b64
//  - async global->LDS staging (ASYNCcnt) via gfx1250 builtin
// ---------------------------------------------------------------------------

typedef __attribute__((ext_vector_type(2))) float v2f;
typedef __attribute__((ext_vector_type(8))) float v8f;

typedef __attribute__((address_space(1))) int as1_int;
typedef __attribute__((address_space(3))) int as3_int;

#define LRELU_SLOPE 0.01f
#define BN_EPSV 1e-5f

#if __has_builtin(__builtin_amdgcn_global_load_async_to_lds_b32)
#define GIN_HAVE_ASYNC_LDS 1
#endif

// ---------------- scatter stage ----------------

// z[i,c] = (1+eps) * h[i,c]   (fused self-term; eps read on-device)
__global__ void gin_pre_init(const float* __restrict__ h,
                             const float* __restrict__ eps,
                             float* __restrict__ z, long long total) {
  long long i = (long long)blockIdx.x * blockDim.x + threadIdx.x;
  if (i < total) z[i] = (1.0f + eps[0]) * h[i];
}

// z[dst[e], :] += h[src[e], :]  — one lane handles 4 channels (float4 gather)
__global__ void gin_scatter_add(const float* __restrict__ h,
                                const int* __restrict__ src,
                                const int* __restrict__ dst,
                                float* __restrict__ z,
                                int E, int C4 /* = C/4 */) {
  long long idx = (long long)blockIdx.x * blockDim.x + threadIdx.x;
  long long total = (long long)E * C4;
  if (idx >= total) return;
  int e = (int)(idx / C4);
  int j = (int)(idx - (long long)e * C4);
  int s = src[e];
  int d = dst[e];
  const float4 v = ((const float4*)h)[(long long)s * C4 + j];
  float* zp = z + ((long long)d * C4 + j) * 4;
  atomicAdd(zp + 0, v.x);
  atomicAdd(zp + 1, v.y);
  atomicAdd(zp + 2, v.z);
  atomicAdd(zp + 3, v.w);
}

// ---------------- fp32 WMMA GEMM with fused epilogue ----------------
//
// out[M,Nn] = epilogue( A[M,K] @ W[K,Nn] + bias )
// epilogue: optional inference BatchNorm (g,be,m,v) then lreluCount x LeakyReLU.
// Block: 256 threads = 8 waves; each wave computes a 16x128 output strip.
// grid.x tiles M by 128 rows, grid.y tiles Nn by 128 cols.
// W chunk is stored transposed in LDS: Wt[n][k], padded row stride 68 floats
// so a wave's 32 B-fragment ds_load_b64s touch 64 distinct banks.

#define GEMM_KC 64           // K-chunk staged in LDS
#define GEMM_LDSK (GEMM_KC + 4)

__global__ __launch_bounds__(256) void gemm_wmma_f32(
    const float* __restrict__ A, int M, int K,
    const float* __restrict__ Wg, int Nn,
    const float* __restrict__ bias,
    const float* __restrict__ g, const float* __restrict__ be,
    const float* __restrict__ m, const float* __restrict__ v,
    int lreluCount, float* __restrict__ out) {
  __shared__ float Wt[128][GEMM_LDSK];  // transposed chunk: Wt[n][k]

  const int tid     = threadIdx.x;
  const int wave    = tid >> 5;
  const int lane    = tid & 31;
  const int half    = lane >> 4;   // which half of the wave (A/B K-pair select)
  const int lane_lo = lane & 15;
  const int colBase = blockIdx.y * 128;
  const long long rowBase = (long long)blockIdx.x * 128 + (long long)wave * 16;

  v8f acc[8];
  {
    v8f zero = {0.f, 0.f, 0.f, 0.f, 0.f, 0.f, 0.f, 0.f};
#pragma unroll
    for (int t = 0; t < 8; ++t) acc[t] = zero;
  }

  // A-fragment row for this lane (16x4 f32 A layout: lanes 0-15 hold K=k0,k0+1;
  // lanes 16-31 hold K=k0+2,k0+3, same M=lane_lo). Clamp row for tail blocks;
  // stores are guarded so clamped rows never leak into valid output.
  long long arow = rowBase + lane_lo;
  if (arow >= M) arow = M - 1;
  const float* __restrict__ Arow = A + arow * (long long)K;

  for (int kb = 0; kb < K; kb += GEMM_KC) {
    // stage W[kb:kb+KC, colBase:colBase+128] transposed into LDS
    for (int idx = tid; idx < GEMM_KC * 128; idx += 256) {
      int kk = idx >> 7;           // K index (coalesced global row)
      int nn = idx & 127;          // N index
#if defined(GIN_HAVE_ASYNC_LDS)
      __builtin_amdgcn_global_load_async_to_lds_b32(
          (as1_int*)(void*)(Wg + (long long)(kb + kk) * Nn + colBase + nn),
          (as3_int*)(void*)&Wt[nn][kk],
          /*offset=*/0, /*cpol=*/0);
#else
      Wt[nn][kk] = Wg[(long long)(kb + kk) * Nn + colBase + nn];
#endif
    }
#if defined(GIN_HAVE_ASYNC_LDS)
#if __has_builtin(__builtin_amdgcn_s_wait_asynccnt)
    __builtin_amdgcn_s_wait_asynccnt(0);
#else
    asm volatile("s_wait_asynccnt 0x0" ::: "memory");
#endif
#endif
    __syncthreads();

    if (kb + GEMM_KC < K) {
      // speculative prefetch of next chunk (global_prefetch_b8)
      __builtin_prefetch(&Wg[(long long)(kb + GEMM_KC) * Nn + colBase + tid], 0, 1);
    }

#pragma unroll 4
    for (int ks = 0; ks < GEMM_KC; ks += 4) {
      const int krow = ks + 2 * half;
      v2f a = *(const v2f*)(Arow + kb + krow);  // 8B-aligned: ks%4==0, 2*half in {0,2}
#pragma unroll
      for (int t = 0; t < 8; ++t) {
        const int n = t * 16 + lane_lo;
        // B 4x16 layout: VGPR r holds K=r+2*half -> contiguous pair in Wt[n][.]
        v2f b = *(const v2f*)&Wt[n][krow];      // single ds_load_b64
        acc[t] = __builtin_amdgcn_wmma_f32_16x16x4_f32(
            /*neg_a=*/false, a, /*neg_b=*/false, b,
            /*c_mod=*/(short)0, acc[t], /*reuse_a=*/false, /*reuse_b=*/false);
      }
    }
    __syncthreads();
  }

  // epilogue: bias + (optional) BN + lreluCount x LeakyReLU, guarded store.
#pragma unroll
  for (int t = 0; t < 8; ++t) {
    const int n = colBase + t * 16 + lane_lo;
    const float bia = bias ? bias[n] : 0.0f;
    float scale = 1.0f, shift = 0.0f;
    if (g) {
      scale = g[n] * rsqrtf(v[n] + BN_EPSV);
      shift = be[n] - m[n] * scale;
    }
#pragma unroll
    for (int r = 0; r < 8; ++r) {
      const long long row = rowBase + r + 8 * half;  // C/D layout: VGPR r -> M=r(+8)
      if (row < M) {
        float z = acc[t][r] + bia;
        z = z * scale + shift;
        for (int q = 0; q < lreluCount; ++q) z = (z >= 0.f) ? z : LRELU_SLOPE * z;
        out[row * (long long)Nn + n] = z;
      }
    }
  }
}

// ---------------- final projection (K x 1) + sigmoid ----------------
__global__ void final_sigmoid(const float* __restrict__ h,
                              const float* __restrict__ Wf,
                              const float* __restrict__ bf,
                              float* __restrict__ out, int Nrows, int K) {
  int i = blockIdx.x * blockDim.x + threadIdx.x;
  if (i >= Nrows) return;
  const float4* hp = (const float4*)(h + (long long)i * K);
  const float4* wp = (const float4*)Wf;
  float acc = 0.f;
#pragma unroll 4
  for (int j = 0; j < K / 4; ++j) {
    float4 a = hp[j];
    float4 b = wp[j];
    acc += a.x * b.x + a.y * b.y + a.z * b.z + a.w * b.w;
  }
  float z = acc + bf[0];
  out[i] = 1.0f / (1.0f + expf(-z));
}

// ---------------------------------------------------------------------------

static inline int ceil_div_i(long long a, long long b) { return (int)((a + b - 1) / b); }

extern "C" void kernel_launch(void* const* d_in, const int* in_sizes, int n_in,
                              void* d_out, int out_size, void* d_ws, size_t ws_size,
                              hipStream_t stream) {
  (void)n_in; (void)out_size; (void)ws_size;

  const float* x    = (const float*)d_in[0];
  const int*   ei   = (const int*)d_in[1];
  // d_in[2] = batch (unused by the reference math)
  const float* eps1 = (const float*)d_in[3];
  const float* W1   = (const float*)d_in[4];
  const float* b1   = (const float*)d_in[5];
  const float* g1   = (const float*)d_in[6];
  const float* be1  = (const float*)d_in[7];
  const float* m1   = (const float*)d_in[8];
  const float* v1   = (const float*)d_in[9];
  const float* epsL = (const float*)d_in[10];
  const float* WL   = (const float*)d_in[11];
  const float* bL   = (const float*)d_in[12];
  const float* gL   = (const float*)d_in[13];
  const float* beL  = (const float*)d_in[14];
  const float* mL   = (const float*)d_in[15];
  const float* vL   = (const float*)d_in[16];
  const float* Wc1  = (const float*)d_in[17];
  const float* bc1  = (const float*)d_in[18];
  const float* Wcl  = (const float*)d_in[19];
  const float* bcl  = (const float*)d_in[20];
  const float* Wf   = (const float*)d_in[21];
  const float* bf   = (const float*)d_in[22];

  const int Cin = 64, C = 128, S = 256, L = 3, Lc = 2;
  const int N = in_sizes[0] / Cin;
  const int E = in_sizes[1] / 2;
  const int* src = ei;       // edge_index[0]
  const int* dst = ei + E;   // edge_index[1]

  float* buf0 = (float*)d_ws;                 // N x 256 fp32
  float* buf1 = buf0 + (size_t)N * S;         // N x 256 fp32

  // ---------- GIN layer 1 (Cin=64 -> C=128) ----------
  {
    const long long tot = (long long)N * Cin;
    gin_pre_init<<<ceil_div_i(tot, 256), 256, 0, stream>>>(x, eps1, buf1, tot);
    const long long stot = (long long)E * (Cin / 4);
    gin_scatter_add<<<ceil_div_i(stot, 256), 256, 0, stream>>>(x, src, dst, buf1, E, Cin / 4);
    dim3 grid(ceil_div_i(N, 128), 1);
    gemm_wmma_f32<<<grid, 256, 0, stream>>>(buf1, N, Cin, W1, C, b1,
                                            g1, be1, m1, v1, /*lrelu*/2, buf0);
  }

  // ---------- GIN layers 2..4 (C -> C) ----------
  for (int l = 0; l < L; ++l) {
    const long long tot = (long long)N * C;
    gin_pre_init<<<ceil_div_i(tot, 256), 256, 0, stream>>>(buf0, epsL + l, buf1, tot);
    const long long stot = (long long)E * (C / 4);
    gin_scatter_add<<<ceil_div_i(stot, 256), 256, 0, stream>>>(buf0, src, dst, buf1, E, C / 4);
    dim3 grid(ceil_div_i(N, 128), 1);
    gemm_wmma_f32<<<grid, 256, 0, stream>>>(buf1, N, C, WL + (size_t)l * C * C, C,
                                            bL + (size_t)l * C,
                                            gL + (size_t)l * C, beL + (size_t)l * C,
                                            mL + (size_t)l * C, vL + (size_t)l * C,
                                            /*lrelu*/2, buf0);
  }

  // ---------- classifier ----------
  {
    dim3 grid(ceil_div_i(N, 128), S / 128);
    // h @ Wc1 + bc1 (no activation)
    gemm_wmma_f32<<<grid, 256, 0, stream>>>(buf0, N, C, Wc1, S, bc1,
                                            nullptr, nullptr, nullptr, nullptr,
                                            /*lrelu*/0, buf1);
    // two S->S layers with LeakyReLU
    gemm_wmma_f32<<<grid, 256, 0, stream>>>(buf1, N, S, Wcl, S, bcl,
                                            nullptr, nullptr, nullptr, nullptr,
                                            /*lrelu*/1, buf0);
    gemm_wmma_f32<<<grid, 256, 0, stream>>>(buf0, N, S, Wcl + (size_t)S * S, S,
                                            bcl + S,
                                            nullptr, nullptr, nullptr, nullptr,
                                            /*lrelu*/1, buf1);
  }

  // ---------- final projection + sigmoid ----------
  final_sigmoid<<<ceil_div_i(N, 256), 256, 0, stream>>>(buf1, Wf, bf, (float*)d_out, N, S);
  (void)Lc;
}